// AttentiveTransformer_20873541058866
// MI455X (gfx1250) — compile-verified
//
#include <hip/hip_runtime.h>
#include <hip/hip_bf16.h>

// Fused AttentiveTransformer for MI455X (gfx1250, wave32):
//   x = feat @ W^T (fp32 WMMA 16x16x4) -> Ghost-BN(128-row chunks) -> * priors -> sparsemax(D=512)
// One workgroup == one ghost batch (128 rows x 512 cols); GEMM result never leaves LDS.
// A tile is DMA'd into LDS by the Tensor Data Mover (tensor_load_to_lds, TENSORcnt),
// priors are prefetched into L2 during the GEMM, output uses non-temporal stores.

typedef __attribute__((ext_vector_type(2))) float        v2f;
typedef __attribute__((ext_vector_type(4))) float        v4f;
typedef __attribute__((ext_vector_type(8))) float        v8f;
typedef __attribute__((ext_vector_type(4))) unsigned int v4u;
typedef __attribute__((ext_vector_type(8))) int          v8i;
typedef __attribute__((ext_vector_type(4))) int          v4i;

#define IN_DIM   128
#define OUT_DIM  512
#define VBSZ     128
#define A_STRIDE 132   // 128 + 4 pad words: conflict-free ds_load_b64 A-fragments (TDM pad fields)
#define Z_STRIDE 516   // 512 + 4 pad words: conflict-free C-tile stores / float4 row loads
#define BN_EPS   1e-5f

__device__ __forceinline__ float wsum32(float v) {
#pragma unroll
  for (int o = 16; o > 0; o >>= 1) v += __shfl_xor(v, o, 32);
  return v;
}
__device__ __forceinline__ float wmax32(float v) {
#pragma unroll
  for (int o = 16; o > 0; o >>= 1) v = fmaxf(v, __shfl_xor(v, o, 32));
  return v;
}

__launch_bounds__(512, 1)
__global__ void attentive_fused_gfx1250(const float* __restrict__ priors,
                                        const float* __restrict__ feat,
                                        const float* __restrict__ W,
                                        const float* __restrict__ gamma,
                                        const float* __restrict__ beta,
                                        float* __restrict__ out) {
  // Z tile (128 x 516 f32 = 258 KB); A staging (128 x 132 f32 = 66 KB) overlaps its start.
  // A is dead before the first Z write (barrier-separated). Total LDS ~262 KB < 320 KB/WGP.
  __shared__ __align__(16) char smem_raw[VBSZ * Z_STRIDE * sizeof(float)];
  __shared__ __align__(16) float sscale[OUT_DIM];
  __shared__ __align__(16) float sshift[OUT_DIM];

  float* Alds = (float*)smem_raw;
  float* Z    = (float*)smem_raw;

  const int t    = threadIdx.x;
  const int wave = t >> 5;
  const int lane = t & 31;
  const int lo   = lane & 15;
  const int hi   = lane >> 4;
  const int r    = wave & 7;     // row-tile group: rows 16r..16r+15
  const int half = wave >> 3;    // column half: cols 256*half..+255
  const int rowbase = 16 * r;
  const int colbase = 256 * half;
  const long grow0 = (long)blockIdx.x * VBSZ;

  // ---- Phase 1: TDM async-tensor DMA of the A chunk (128x128 f32) into LDS ----
  // D# describes a 128x128 2D tile, data_size=4B, with LDS padding of 4 dwords after
  // every 128 dwords -> row stride 132 floats (matches A_STRIDE). Issued once (wave 0);
  // TDM ignores EXEC and signals completion via TENSORcnt.
#if __has_builtin(__builtin_amdgcn_tensor_load_to_lds)
  if (wave == 0) {
    unsigned long long ga = (unsigned long long)(const void*)(feat + grow0 * IN_DIM);
    unsigned lds_off = (unsigned)(unsigned long long)(void*)smem_raw;  // generic addr low 32b == LDS offset
    v4u g0;
    g0[0] = 1u;                                          // count=1, user-mode descriptor
    g0[1] = lds_off;                                     // lds_addr
    g0[2] = (unsigned)(ga & 0xFFFFFFFFu);                // global_addr[31:0]
    g0[3] = (unsigned)((ga >> 32) & 0x01FFFFFFu)         // global_addr[56:32]
            | (2u << 30);                                // type = 2 ("image")
    v8i g1;
    g1[0] = (int)((2u << 16)        // data_size = 4 bytes
                  | (1u << 20)      // pad_enable
                  | (6u << 22)      // pad_interval: 128 dwords
                  | (3u << 25));    // pad_amount: 4 dwords
    g1[1] = (int)(128u << 16);      // tensor_dim0 = 128 (bits 79:48, low half)
    g1[2] = (int)(128u << 16);      // tensor_dim0 hi=0 | tensor_dim1 = 128 (low half)
    g1[3] = (int)(128u << 16);      // tensor_dim1 hi=0 | tile_dim0 = 128
    g1[4] = (int)128u;              // tile_dim1 = 128 | tile_dim2 = 0
    g1[5] = (int)128u;              // tensor_dim0_stride = 128 elements
    g1[6] = 0;                      // stride hi | tensor_dim1_stride lo (unused, 2D)
    g1[7] = 0;
    v4i gz4 = {0, 0, 0, 0};         // groups 2/3 unused for 2D tiles
    v8i gz8 = {0, 0, 0, 0, 0, 0, 0, 0};
    __builtin_amdgcn_tensor_load_to_lds(g0, g1, gz4, gz4, gz8, 0);
  }
#else
  {
    const float4* src = (const float4*)(feat + grow0 * IN_DIM);
#pragma unroll
    for (int i = 0; i < 8; ++i) {
      int fidx = i * 512 + t;
      int row  = fidx >> 5;
      int c4   = fidx & 31;
      float4 v = src[fidx];
      *(float4*)&Alds[row * A_STRIDE + c4 * 4] = v;
    }
  }
#endif

  // Prefetch this workgroup's priors slab (256 KB) into L2 while the DMA + GEMM run.
  // global_prefetch_b8 does not touch LOADcnt -> fire-and-forget.
  {
    const float* pbase = priors + grow0 * OUT_DIM;
#pragma unroll
    for (int i = 0; i < 4; ++i)
      __builtin_prefetch(pbase + (t + i * 512) * 32, 0, 3);  // one 128B line per call
  }

#if __has_builtin(__builtin_amdgcn_tensor_load_to_lds)
  if (wave == 0) __builtin_amdgcn_s_wait_tensorcnt(0);
#endif
  __syncthreads();

  // ---- Phase 2: GEMM via V_WMMA_F32_16X16X4_F32 ----
  // A frag (16x4): lanes 0-15 hold M=lo, {K=k0,k0+1}; lanes 16-31 hold {K=k0+2,k0+3}.
  // B frag (4x16): lane group picks K=k0+2*hi, N=colbase+16c+lo; B[k][n] = W[n][k] (row-major W).
  v8f acc[16];
#pragma unroll
  for (int c = 0; c < 16; ++c) acc[c] = v8f{};

  for (int k0 = 0; k0 < IN_DIM; k0 += 4) {
    v2f a = *(const v2f*)&Alds[(rowbase + lo) * A_STRIDE + k0 + 2 * hi];
#pragma unroll
    for (int c = 0; c < 16; ++c) {
      const float* wrow = W + (colbase + c * 16 + lo) * IN_DIM + (k0 + 2 * hi);
      v2f b = *(const v2f*)wrow;  // W is tiny (256 KB): L2-resident, reused by all 1024 blocks
      acc[c] = __builtin_amdgcn_wmma_f32_16x16x4_f32(
          /*neg_a=*/false, a, /*neg_b=*/false, b,
          /*c_mod=*/(short)0, acc[c], /*reuse_a=*/false, /*reuse_b=*/false);
    }
  }
  __syncthreads();  // all A reads done before Z overwrites the union

  // ---- Phase 3: spill C tiles to LDS Z (VGPR j -> M = j + 8*hi, N = lane%16) ----
#pragma unroll
  for (int c = 0; c < 16; ++c) {
#pragma unroll
    for (int j = 0; j < 8; ++j) {
      Z[(rowbase + j + 8 * hi) * Z_STRIDE + colbase + c * 16 + lo] = acc[c][j];
    }
  }
  __syncthreads();

  // ---- Phase 4: ghost-BN stats: one thread per column (bank-conflict-free) ----
  {
    const int n = t;
    float s = 0.f, ss = 0.f;
#pragma unroll 4
    for (int m = 0; m < VBSZ; ++m) {
      float v = Z[m * Z_STRIDE + n];
      s += v;
      ss = fmaf(v, v, ss);
    }
    float mean = s * (1.0f / VBSZ);
    float var  = fmaf(-mean, mean, ss * (1.0f / VBSZ));
    float rs   = rsqrtf(var + BN_EPS);
    float sc   = gamma[n] * rs;
    sscale[n]  = sc;
    sshift[n]  = fmaf(-mean, sc, beta[n]);
  }
  __syncthreads();

  // ---- Phase 5: BN apply + priors + sparsemax; one wave per row, 8 rows/wave ----
  for (int i = 0; i < 8; ++i) {
    const int  m    = wave * 8 + i;
    const long grow = grow0 + m;
    float x[16];
#pragma unroll
    for (int j = 0; j < 4; ++j) {
      const int col = lane * 4 + j * 128;
      v4f zv = *(const v4f*)&Z[m * Z_STRIDE + col];
      v4f sc = *(const v4f*)&sscale[col];
      v4f sh = *(const v4f*)&sshift[col];
      v4f pr = *(const v4f*)(priors + grow * OUT_DIM + col);  // fully coalesced, L2-prefetched
#pragma unroll
      for (int q = 0; q < 4; ++q)
        x[j * 4 + q] = fmaf(zv[q], sc[q], sh[q]) * pr[q];
    }

    // shift by row max
    float vmax = x[0];
#pragma unroll
    for (int q = 1; q < 16; ++q) vmax = fmaxf(vmax, x[q]);
    vmax = wmax32(vmax);
#pragma unroll
    for (int q = 0; q < 16; ++q) x[q] -= vmax;

    // sparsemax tau via Michelot fixed-point (exact at convergence; tau monotone increasing)
    float s0 = 0.f;
#pragma unroll
    for (int q = 0; q < 16; ++q) s0 += x[q];
    float tau = (wsum32(s0) - 1.0f) * (1.0f / OUT_DIM);
    for (int it = 0; it < 16; ++it) {
      float s = 0.f, k = 0.f;
#pragma unroll
      for (int q = 0; q < 16; ++q) {
        bool gt = x[q] > tau;
        s += gt ? x[q] : 0.0f;
        k += gt ? 1.0f : 0.0f;
      }
      s = wsum32(s);
      k = wsum32(k);             // k >= 1 always (max element stays in support)
      tau = (s - 1.0f) / k;
    }

    // clip(x - tau, 0) -> coalesced non-temporal v4f stores (write-once stream)
#pragma unroll
    for (int j = 0; j < 4; ++j) {
      const int col = lane * 4 + j * 128;
      v4f o;
#pragma unroll
      for (int q = 0; q < 4; ++q)
        o[q] = fmaxf(x[j * 4 + q] - tau, 0.0f);
      __builtin_nontemporal_store(o, (v4f*)(out + grow * OUT_DIM + col));
    }
  }
}

extern "C" void kernel_launch(void* const* d_in, const int* in_sizes, int n_in,
                              void* d_out, int out_size, void* d_ws, size_t ws_size,
                              hipStream_t stream) {
  const float* priors = (const float*)d_in[0];
  const float* feat   = (const float*)d_in[1];
  const float* W      = (const float*)d_in[2];
  const float* gamma  = (const float*)d_in[3];
  const float* beta   = (const float*)d_in[4];
  float* out = (float*)d_out;

  const int B       = in_sizes[1] / IN_DIM;   // 131072
  const int nchunks = B / VBSZ;               // 1024 workgroups, one ghost batch each

  attentive_fused_gfx1250<<<nchunks, 512, 0, stream>>>(priors, feat, W, gamma, beta, out);
}